// PotentialLoss_47210280517636
// MI455X (gfx1250) — compile-verified
//
#include <hip/hip_runtime.h>
#include <math.h>

// Problem constants (match reference)
#define N_HITS  32768
#define P_PART  2048
#define D_DIM   16
#define Q_MIN   0.01f
#define RADIUS  1.0f
#define PT_THLD 0.9f

typedef __attribute__((ext_vector_type(2))) float v2f;
typedef __attribute__((ext_vector_type(8))) float v8f;

// ---- workspace layout (bytes) ----
#define WS_KEY   0                           // u64[P]   : packed (q_bits<<32)|~idx argmax keys
#define WS_Q     (WS_KEY  + P_PART * 8)      // f32[N]   : q per hit
#define WS_NH    (WS_Q    + N_HITS * 4)      // f32[N]   : ||x_i||^2
#define WS_MASK  (WS_NH   + N_HITS * 4)      // f32[N]   : mask as 0/1
#define WS_XA    (WS_MASK + N_HITS * 4)      // f32[P*16]: gathered x_alpha rows
#define WS_QA    (WS_XA   + P_PART * D_DIM * 4) // f32[P]: q_alpha
#define WS_NPA   (WS_QA   + P_PART * 4)      // f32[P]   : ||x_alpha||^2
#define WS_NMASK (WS_NPA  + P_PART * 4)      // u32      : n_masked
#define WS_PART  (WS_NMASK + 8)              // f32[2*GRID3] partial sums

#define GRID3  256   // N_HITS / (8 waves * 16 rows)
#define BLK3   256
#define SUPER  256   // particles staged in LDS per super-tile
#define ROWPAD 20    // floats per staged xa row (16B-aligned, bank-conflict-free)

__global__ void k_init(unsigned char* ws) {
    int i = blockIdx.x * 256 + threadIdx.x;
    unsigned long long* key = (unsigned long long*)(ws + WS_KEY);
    if (i < P_PART) key[i] = 0ull;
    if (i == 0) *(unsigned int*)(ws + WS_NMASK) = 0u;
}

__global__ void k_prep(const float* __restrict__ beta, const float* __restrict__ x,
                       const int* __restrict__ pid, const int* __restrict__ recon,
                       const float* __restrict__ pt, unsigned char* ws) {
    int i = blockIdx.x * 256 + threadIdx.x;
    if (i >= N_HITS) return;
    unsigned long long* key = (unsigned long long*)(ws + WS_KEY);
    float* q  = (float*)(ws + WS_Q);
    float* nh = (float*)(ws + WS_NH);
    float* mk = (float*)(ws + WS_MASK);
    unsigned int* nm = (unsigned int*)(ws + WS_NMASK);

    float a  = atanhf(beta[i]);
    float qi = a * a + Q_MIN;          // qi > 0 always => float bits are order-preserving
    q[i] = qi;

    float s = 0.f;
#pragma unroll
    for (int d = 0; d < D_DIM; ++d) { float v = x[i * D_DIM + d]; s += v * v; }
    nh[i] = s;

    bool m = (recon[i] > 0) && (pt[i] > PT_THLD);
    mk[i] = m ? 1.f : 0.f;
    if (m) atomicAdd(nm, 1u);

    int p = pid[i];
    if (p >= 1 && p <= P_PART) {
        unsigned long long k = ((unsigned long long)__float_as_uint(qi) << 32)
                             | (unsigned int)(~(unsigned int)i);   // tie -> smaller index wins
        atomicMax(&key[p - 1], k);
    }
}

__global__ void k_gather(const float* __restrict__ x, unsigned char* ws) {
    int j = blockIdx.x * 256 + threadIdx.x;
    if (j >= P_PART) return;
    const unsigned long long* key = (const unsigned long long*)(ws + WS_KEY);
    const float* q  = (const float*)(ws + WS_Q);
    const float* nh = (const float*)(ws + WS_NH);
    float* xa  = (float*)(ws + WS_XA);
    float* qa  = (float*)(ws + WS_QA);
    float* npa = (float*)(ws + WS_NPA);

    unsigned long long k = key[j];
    unsigned int alpha = (k == 0ull) ? 0u : ~(unsigned int)(k & 0xFFFFFFFFull);
    if (alpha >= N_HITS) alpha = 0u;   // matches argmax-of-zeros -> index 0
#pragma unroll
    for (int d = 0; d < D_DIM; ++d) xa[j * D_DIM + d] = x[alpha * D_DIM + d];
    qa[j]  = q[alpha];
    npa[j] = nh[alpha];
}

// Main pairwise kernel: 8 waves/block, each wave owns a 16-hit row block.
// xa/qa/npa are staged in LDS in 256-particle super-tiles shared by all waves.
// Dot products via chained V_WMMA_F32_16X16X4_F32 with A pre-scaled by -2 so
// the accumulator directly contributes -2*dot to d2.
__global__ void __launch_bounds__(BLK3) k_pairs(const float* __restrict__ x,
                                                const int* __restrict__ pid,
                                                unsigned char* ws) {
    const int lane = threadIdx.x & 31;
    const int wave = threadIdx.x >> 5;
    const int hit0 = blockIdx.x * 128 + wave * 16;

    const float* qws = (const float*)(ws + WS_Q);
    const float* nh  = (const float*)(ws + WS_NH);
    const float* mk  = (const float*)(ws + WS_MASK);
    const float* xa  = (const float*)(ws + WS_XA);
    const float* qa  = (const float*)(ws + WS_QA);
    const float* npa = (const float*)(ws + WS_NPA);

    __shared__ float sXA[SUPER * ROWPAD];   // 20 KB, rows 16B-aligned
    __shared__ float sQA[SUPER];
    __shared__ float sNPA[SUPER];

    const int r16 = lane & 15;
    const int hi  = lane >> 4;   // lane half
    const int h   = hi * 2;      // dim offset within each K-group of 4

    // A fragments (ISA 32-bit 16x4 layout, 4 chained K-steps), pre-scaled by -2.
    float af[8];
#pragma unroll
    for (int k = 0; k < 4; ++k) {
        const float* row = x + (size_t)(hit0 + r16) * D_DIM;
        af[2 * k]     = -2.f * row[4 * k + h];
        af[2 * k + 1] = -2.f * row[4 * k + h + 1];
    }

    // Epilogue per-row data: C/D layout -> lane L, acc[r] <-> row M = r + 8*hi
    float qh[8], qm[8], nhh[8];
    int pidm[8];
#pragma unroll
    for (int r = 0; r < 8; ++r) {
        int row = hit0 + r + 8 * hi;
        qh[r]   = qws[row];
        qm[r]   = qh[r] * mk[row];
        nhh[r]  = nh[row];
        pidm[r] = pid[row];
    }

    float att = 0.f, rep = 0.f;

    for (int super = 0; super < P_PART; super += SUPER) {
        // Cooperative stage of one super-tile: thread t copies xa row (super+t)
        {
            int j = super + threadIdx.x;
            const float4* s4 = (const float4*)(xa + (size_t)j * D_DIM);
            float4* d4 = (float4*)(sXA + threadIdx.x * ROWPAD);
            d4[0] = s4[0]; d4[1] = s4[1]; d4[2] = s4[2]; d4[3] = s4[3];
            sQA[threadIdx.x]  = qa[j];
            sNPA[threadIdx.x] = npa[j];
        }
        __syncthreads();

        for (int t = 0; t < SUPER; t += 16) {
            const int   colL = t + r16;
            const float qac  = sQA[colL];
            const float npac = sNPA[colL];
            const int   pidc = super + colL + 1;   // pid value of this column

            // B fragments (4x16 KxN layout, lane-symmetric twin of A) from LDS.
            float bf[8];
#pragma unroll
            for (int k = 0; k < 4; ++k) {
                const float* row = sXA + colL * ROWPAD;
                bf[2 * k]     = row[4 * k + h];
                bf[2 * k + 1] = row[4 * k + h + 1];
            }

            v8f acc = {};
#pragma unroll
            for (int k = 0; k < 4; ++k) {
                v2f a = { af[2 * k], af[2 * k + 1] };
                v2f b = { bf[2 * k], bf[2 * k + 1] };
                acc = __builtin_amdgcn_wmma_f32_16x16x4_f32(
                    /*neg_a=*/false, a, /*neg_b=*/false, b,
                    /*c_mod=*/(short)0, acc, /*reuse_a=*/false, /*reuse_b=*/false);
            }

            // Branchless epilogue: acc[r] == -2*dot already.
            float attT = 0.f, repT = 0.f;
#pragma unroll
            for (int r = 0; r < 8; ++r) {
                float d2   = fmaxf(nhh[r] + npac + acc[r], 0.f);
                float dist = __builtin_amdgcn_sqrtf(d2);       // single v_sqrt_f32
                float rt   = fmaxf(RADIUS - dist, 0.f);
                bool  same = (pidm[r] == pidc);
                attT += same ? qm[r] * d2 : 0.f;
                repT += same ? 0.f : qh[r] * rt;
            }
            att = fmaf(qac, attT, att);
            rep = fmaf(qac, repT, rep);
        }
        __syncthreads();
    }

    // Block reduction -> per-block partials (deterministic)
    __shared__ float sA[BLK3];
    __shared__ float sR[BLK3];
    sA[threadIdx.x] = att;
    sR[threadIdx.x] = rep;
    __syncthreads();
    for (int s = BLK3 / 2; s > 0; s >>= 1) {
        if ((int)threadIdx.x < s) {
            sA[threadIdx.x] += sA[threadIdx.x + s];
            sR[threadIdx.x] += sR[threadIdx.x + s];
        }
        __syncthreads();
    }
    if (threadIdx.x == 0) {
        float* part = (float*)(ws + WS_PART);
        part[blockIdx.x]         = sA[0];
        part[GRID3 + blockIdx.x] = sR[0];
    }
}

__global__ void k_final(unsigned char* ws, float* out) {
    __shared__ float sA[GRID3];
    __shared__ float sR[GRID3];
    const float* part = (const float*)(ws + WS_PART);
    int t = threadIdx.x;
    sA[t] = part[t];
    sR[t] = part[GRID3 + t];
    __syncthreads();
    for (int s = GRID3 / 2; s > 0; s >>= 1) {
        if (t < s) { sA[t] += sA[t + s]; sR[t] += sR[t + s]; }
        __syncthreads();
    }
    if (t == 0) {
        unsigned int nm = *(const unsigned int*)(ws + WS_NMASK);
        out[0] = sA[0] / (float)nm;
        out[1] = sR[0] / (float)N_HITS;
    }
}

extern "C" void kernel_launch(void* const* d_in, const int* in_sizes, int n_in,
                              void* d_out, int out_size, void* d_ws, size_t ws_size,
                              hipStream_t stream) {
    const float* beta  = (const float*)d_in[0];
    const float* x     = (const float*)d_in[1];
    const int*   pidv  = (const int*)d_in[2];
    const int*   recon = (const int*)d_in[3];
    const float* pt    = (const float*)d_in[4];
    // d_in[5] = n_particles scalar; P is compile-time constant here.
    unsigned char* ws = (unsigned char*)d_ws;
    float* out = (float*)d_out;

    k_init  <<<(P_PART + 255) / 256, 256, 0, stream>>>(ws);
    k_prep  <<<N_HITS / 256,        256, 0, stream>>>(beta, x, pidv, recon, pt, ws);
    k_gather<<<P_PART / 256,        256, 0, stream>>>(x, ws);
    k_pairs <<<GRID3,              BLK3, 0, stream>>>(x, pidv, ws);
    k_final <<<1,                 GRID3, 0, stream>>>(ws, out);
}